// SubWindow_49160195670113
// MI455X (gfx1250) — compile-verified
//
#include <hip/hip_runtime.h>
#include <hip/hip_bf16.h>

typedef __attribute__((ext_vector_type(2))) float v2f;
typedef __attribute__((ext_vector_type(8))) float v8f;

#define IMG_B 16
#define IMG_C 3
#define IMG_H 720
#define IMG_W 1280
#define OUTS  255
#define IMG_ELEMS (IMG_C * IMG_H * IMG_W)     // 2,764,800 floats per image
#define NB_PER_IMG 300                        // blocks per image for the mean pass
#define ITERS 9                               // 300 * 9 * (256 threads * 4 floats) = 2,764,800

// ---------------------------------------------------------------------------
// Kernel 1: per-block partial sums of each image.
// Each thread accumulates 9 coalesced float4 loads (36 values), then the
// wave's 32 lane-partials are collapsed with a single V_WMMA_F32_16X16X4_F32:
//   A[m][k] holds the lane partials (lane L<16 -> A[L][0], lane L>=16 -> A[L-16][2]),
//   B = all-ones  =>  D[m][n] = sum_k A[m][k]  (row sums, replicated over n).
// Summing c[0..7] in lane N gives rows 0..7 (lanes 0-15) / rows 8..15 (lanes 16-31),
// so wave_total = t(lane 0) + t(lane 16). LDS combines the 8 waves of the block.
// ---------------------------------------------------------------------------
__global__ void mean_partial_kernel(const float* __restrict__ im,
                                    float* __restrict__ partials) {
    const int blk = blockIdx.x;               // 0 .. 16*300-1
    const int img = blk / NB_PER_IMG;
    const int sub = blk % NB_PER_IMG;
    const int tid = threadIdx.x;              // 256 threads = 8 wave32

    const size_t base = (size_t)img * IMG_ELEMS + (size_t)sub * (ITERS * 1024);
    const float4* p = reinterpret_cast<const float4*>(im + base) + tid;

    float s = 0.0f;
#pragma unroll
    for (int i = 0; i < ITERS; ++i) {
        float4 v = p[i * 256];
        s += (v.x + v.y) + (v.z + v.w);
    }

    // Cross-lane reduction via WMMA (EXEC is all-ones here: no divergence yet).
    v2f a; a[0] = s;    a[1] = 0.0f;
    v2f b; b[0] = 1.0f; b[1] = 1.0f;
    v8f c = {};
    c = __builtin_amdgcn_wmma_f32_16x16x4_f32(
            /*neg_a=*/false, a, /*neg_b=*/false, b,
            /*c_mod=*/(short)0, c, /*reuse_a=*/false, /*reuse_b=*/false);

    float t = ((c[0] + c[1]) + (c[2] + c[3])) + ((c[4] + c[5]) + (c[6] + c[7]));

    __shared__ float red[16];                 // 8 waves * 2 half-wave totals
    const int lane = tid & 31;
    const int wave = tid >> 5;
    if (lane == 0 || lane == 16)
        red[wave * 2 + (lane >> 4)] = t;
    __syncthreads();

    if (tid == 0) {
        float tot = 0.0f;
#pragma unroll
        for (int i = 0; i < 16; ++i) tot += red[i];
        partials[blk] = tot;
    }
}

// ---------------------------------------------------------------------------
// Kernel 2: reduce the 300 partials of each image -> mean.
// ---------------------------------------------------------------------------
__global__ void mean_finalize_kernel(const float* __restrict__ partials,
                                     float* __restrict__ means) {
    const int img = blockIdx.x;               // 16 blocks
    const int tid = threadIdx.x;              // 256 threads
    __shared__ float sm[256];

    float s = 0.0f;
    for (int i = tid; i < NB_PER_IMG; i += 256)
        s += partials[img * NB_PER_IMG + i];
    sm[tid] = s;
    __syncthreads();
    for (int off = 128; off > 0; off >>= 1) {
        if (tid < off) sm[tid] += sm[tid + off];
        __syncthreads();
    }
    if (tid == 0)
        means[img] = sm[0] * (1.0f / (float)IMG_ELEMS);
}

// ---------------------------------------------------------------------------
// Kernel 3: fused crop + mean-pad + bilinear resize (align_corners=False).
// One block per (image, channel, output row); thread x = output column.
// rintf == jnp.round (round-half-even); addresses are clipped so every load
// is safe, and invalid taps are replaced by the image mean.
// ---------------------------------------------------------------------------
__global__ void crop_resize_kernel(const float* __restrict__ im,
                                   const float* __restrict__ pos,
                                   const float* __restrict__ size_wh,
                                   const float* __restrict__ means,
                                   float* __restrict__ out) {
    const int bidx = blockIdx.x;              // 16*3*255 blocks
    const int y  = bidx % OUTS;
    const int bc = bidx / OUTS;
    const int ch = bc % IMG_C;
    const int b  = bc / IMG_C;
    const int x  = threadIdx.x;
    if (x >= OUTS) return;

    const float sz  = size_wh[b];
    const float px  = pos[b * 2 + 0];
    const float py  = pos[b * 2 + 1];
    const float avg = means[b];

    const float half  = (sz + 1.0f) * 0.5f;
    const float xmin  = rintf(px - half);
    const float ymin  = rintf(py - half);
    const float scale = sz / (float)OUTS;
    const float szm1  = sz - 1.0f;

    const float srcy = fminf(fmaxf(((float)y + 0.5f) * scale - 0.5f, 0.0f), szm1);
    const float loy  = floorf(srcy);
    const float wy   = srcy - loy;
    const float hiy  = fminf(loy + 1.0f, szm1);

    const float srcx = fminf(fmaxf(((float)x + 0.5f) * scale - 0.5f, 0.0f), szm1);
    const float lox  = floorf(srcx);
    const float wx   = srcx - lox;
    const float hix  = fminf(lox + 1.0f, szm1);

    const float y0f = loy + ymin, y1f = hiy + ymin;
    const float x0f = lox + xmin, x1f = hix + xmin;

    const float* plane = im + ((size_t)b * IMG_C + ch) * (IMG_H * IMG_W);

    auto sample = [&](float yf, float xf) -> float {
        const bool valid = (yf >= 0.0f) && (yf <= (float)(IMG_H - 1)) &&
                           (xf >= 0.0f) && (xf <= (float)(IMG_W - 1));
        const int yc = (int)fminf(fmaxf(yf, 0.0f), (float)(IMG_H - 1));
        const int xc = (int)fminf(fmaxf(xf, 0.0f), (float)(IMG_W - 1));
        const float v = plane[(size_t)yc * IMG_W + xc];
        return valid ? v : avg;
    };

    const float v00 = sample(y0f, x0f);
    const float v01 = sample(y0f, x1f);
    const float v10 = sample(y1f, x0f);
    const float v11 = sample(y1f, x1f);

    const float top = v00 * (1.0f - wx) + v01 * wx;
    const float bot = v10 * (1.0f - wx) + v11 * wx;
    const float r   = top * (1.0f - wy) + bot * wy;

    out[(((size_t)b * IMG_C + ch) * OUTS + y) * OUTS + x] = r;
}

// ---------------------------------------------------------------------------
extern "C" void kernel_launch(void* const* d_in, const int* in_sizes, int n_in,
                              void* d_out, int out_size, void* d_ws, size_t ws_size,
                              hipStream_t stream) {
    const float* im      = (const float*)d_in[0];   // (16,3,720,1280) f32
    const float* pos     = (const float*)d_in[1];   // (16,2) f32
    const float* size_wh = (const float*)d_in[2];   // (16,) f32
    float*       out     = (float*)d_out;           // (16,1,3,255,255) f32

    float* partials = (float*)d_ws;                 // 16*300 floats
    float* means    = partials + IMG_B * NB_PER_IMG;// 16 floats

    mean_partial_kernel<<<IMG_B * NB_PER_IMG, 256, 0, stream>>>(im, partials);
    mean_finalize_kernel<<<IMG_B, 256, 0, stream>>>(partials, means);
    crop_resize_kernel<<<IMG_B * IMG_C * OUTS, 256, 0, stream>>>(im, pos, size_wh, means, out);
}